// GCN_traffic_1219770712262
// MI455X (gfx1250) — compile-verified
//
#include <hip/hip_runtime.h>

#define N_NODES 50000
#define N_EDGES 800000
#define FDIM    256
#define NGR     64
#define MTILES  (N_NODES / 16)   // 3125

typedef __attribute__((ext_vector_type(2))) float v2f;
typedef __attribute__((ext_vector_type(4))) float v4f;
typedef __attribute__((ext_vector_type(8))) float v8f;

// ---------------- degree / normalization ----------------
__global__ void deg_init(float* deg) {
    int i = blockIdx.x * blockDim.x + threadIdx.x;
    if (i < N_NODES) deg[i] = 1.0f;              // self-loop contributes 1
}

__global__ void deg_count(const int* __restrict__ esrc, float* deg) {
    int e = blockIdx.x * blockDim.x + threadIdx.x;
    if (e < N_EDGES) atomicAdd(&deg[esrc[e]], 1.0f);   // dst2 = edge_src in reversed graph
}

__global__ void deg_rsqrt(float* deg) {
    int i = blockIdx.x * blockDim.x + threadIdx.x;
    if (i < N_NODES) deg[i] = rsqrtf(deg[i]);    // deg >= 1 always
}

__global__ void zero_out_k(float* out) {
    int i = blockIdx.x * blockDim.x + threadIdx.x;
    if (i < NGR * FDIM) out[i] = 0.0f;
}

// ---------------- dense GEMM via fp32 WMMA ----------------
// One wave computes a 16-row strip across all 256 output columns
// (16 accumulator tiles of v8f). W staged through LDS in 32-K-row chunks.
__launch_bounds__(128, 1)
__global__ void gcn_gemm_wmma(const float* __restrict__ H,
                              const float* __restrict__ W,
                              float* __restrict__ T) {
    __shared__ float lw[32 * FDIM];              // 32 KB chunk of W
    const int tid  = threadIdx.x;
    const int lane = tid & 31;

    int mtile = blockIdx.x * 4 + (tid >> 5);
    if (mtile >= MTILES) mtile = MTILES - 1;     // clamp: keep EXEC all-1s for WMMA

    const int mrow = (mtile << 4) + (lane & 15);
    const int koff = (lane >> 4) << 1;           // 0 (lanes 0-15) or 2 (lanes 16-31)
    const int ncol = lane & 15;
    const float* arow = H + (size_t)mrow * FDIM + koff;

    v8f acc[16];
#pragma unroll
    for (int t = 0; t < 16; ++t) acc[t] = (v8f)0.0f;

    for (int kc = 0; kc < FDIM; kc += 32) {
        __syncthreads();
        // stage W[kc..kc+31][0..255] into LDS, coalesced float4, conflict-free
#pragma unroll
        for (int i = 0; i < 16; ++i) {
            const int idx = (i * 128 + tid) * 4;
            *(v4f*)&lw[idx] = *(const v4f*)&W[(size_t)kc * FDIM + idx];
        }
        __syncthreads();
#pragma unroll
        for (int k2 = 0; k2 < 32; k2 += 4) {
            // A 16x4 layout: lanes 0-15 hold K=k2,k2+1; lanes 16-31 hold K=k2+2,k2+3
            const v2f a = *(const v2f*)(arow + kc + k2);
#pragma unroll
            for (int t = 0; t < 16; ++t) {
                // B 4x16 layout: VGPR0 = rows K (lanes 0-15) / K+2 (lanes 16-31), VGPR1 = K+1 / K+3
                v2f b;
                b.x = lw[(k2 + koff    ) * FDIM + t * 16 + ncol];
                b.y = lw[(k2 + koff + 1) * FDIM + t * 16 + ncol];
                acc[t] = __builtin_amdgcn_wmma_f32_16x16x4_f32(
                    false, a, false, b, (short)0, acc[t], false, false);
            }
        }
    }

    // D layout: VGPR g holds M = g + (lane<16 ? 0 : 8), N = lane & 15
    const int mbase = (mtile << 4) + ((lane >> 4) << 3);
#pragma unroll
    for (int t = 0; t < 16; ++t) {
#pragma unroll
        for (int g = 0; g < 8; ++g) {
            T[(size_t)(mbase + g) * FDIM + t * 16 + ncol] = acc[t][g];
        }
    }
}

// ---------------- self-loop init: A = T * dinv[i]^2 ----------------
__global__ void self_init(const float* __restrict__ T, const float* __restrict__ dinv,
                          float* __restrict__ A) {
    int i = blockIdx.x * blockDim.x + threadIdx.x;     // over N*64 float4 chunks
    if (i >= N_NODES * 64) return;
    int node = i >> 6;
    float w = dinv[node]; w *= w;
    v4f v = *(const v4f*)&T[(size_t)i * 4];
    *(v4f*)&A[(size_t)i * 4] = v * w;
}

// ---------------- edge scatter-add (L2-resident atomics) ----------------
__global__ void edge_scatter(const float* __restrict__ T, const float* __restrict__ dinv,
                             const int* __restrict__ esrc, const int* __restrict__ edst,
                             float* __restrict__ A) {
    long long i = (long long)blockIdx.x * blockDim.x + threadIdx.x;  // E*64 chunks
    if (i >= (long long)N_EDGES * 64) return;
    int e = (int)(i >> 6);
    int c = ((int)i & 63) << 2;
    int s = edst[e];                 // message source in reversed graph
    int d = esrc[e];                 // message target
    float w = dinv[s] * dinv[d];
    v4f v = *(const v4f*)&T[(size_t)s * FDIM + c] * w;
    float* dst = &A[(size_t)d * FDIM + c];
    atomicAdd(dst + 0, v.x);
    atomicAdd(dst + 1, v.y);
    atomicAdd(dst + 2, v.z);
    atomicAdd(dst + 3, v.w);
}

// ---------------- epilogue: H' = relu(A + b) ----------------
__global__ void bias_relu(const float* __restrict__ A, const float* __restrict__ b,
                          float* __restrict__ Hn) {
    int i = blockIdx.x * blockDim.x + threadIdx.x;
    if (i >= N_NODES * 64) return;
    int c = (i & 63) << 2;
    v4f bb = *(const v4f*)&b[c];
    v4f v = *(const v4f*)&A[(size_t)i * 4] + bb;
    v.x = fmaxf(v.x, 0.0f); v.y = fmaxf(v.y, 0.0f);
    v.z = fmaxf(v.z, 0.0f); v.w = fmaxf(v.w, 0.0f);
    *(v4f*)&Hn[(size_t)i * 4] = v;
}

// ---------------- final layer: pool(A + bo) by batch ----------------
__global__ void pool_add(const float* __restrict__ A, const float* __restrict__ bo,
                         const int* __restrict__ batch, float* __restrict__ out) {
    int i = blockIdx.x * blockDim.x + threadIdx.x;
    if (i >= N_NODES * 64) return;
    int node = i >> 6;
    int c = (i & 63) << 2;
    int g = batch[node];
    v4f v = *(const v4f*)&A[(size_t)i * 4] + *(const v4f*)&bo[c];
    float* dst = &out[(size_t)g * FDIM + c];
    atomicAdd(dst + 0, v.x);
    atomicAdd(dst + 1, v.y);
    atomicAdd(dst + 2, v.z);
    atomicAdd(dst + 3, v.w);
}

extern "C" void kernel_launch(void* const* d_in, const int* in_sizes, int n_in,
                              void* d_out, int out_size, void* d_ws, size_t ws_size,
                              hipStream_t stream) {
    const float* x     = (const float*)d_in[0];
    const int*   esrc  = (const int*)d_in[1];
    const int*   edst  = (const int*)d_in[2];
    const int*   batch = (const int*)d_in[3];
    const float* Wm[4] = {(const float*)d_in[4], (const float*)d_in[6],
                          (const float*)d_in[8], (const float*)d_in[10]};
    const float* bm[4] = {(const float*)d_in[5], (const float*)d_in[7],
                          (const float*)d_in[9], (const float*)d_in[11]};
    float* out = (float*)d_out;

    char* ws = (char*)d_ws;
    float* dinv = (float*)ws;
    size_t off  = ((size_t)N_NODES * 4 + 255) & ~(size_t)255;
    float* buf1 = (float*)(ws + off);
    float* buf2 = (float*)(ws + off + (size_t)N_NODES * FDIM * 4);

    dim3 b256(256);
    deg_init <<<(N_NODES + 255) / 256, b256, 0, stream>>>(dinv);
    deg_count<<<(N_EDGES + 255) / 256, b256, 0, stream>>>(esrc, dinv);
    deg_rsqrt<<<(N_NODES + 255) / 256, b256, 0, stream>>>(dinv);
    zero_out_k<<<(NGR * FDIM + 255) / 256, b256, 0, stream>>>(out);

    const int NT = N_NODES * 64;
    const long long ET = (long long)N_EDGES * 64;
    const int gemm_blocks = (MTILES + 3) / 4;
    const int eblk = (int)((ET + 255) / 256);

    const float* H = x;
    for (int l = 0; l < 4; ++l) {
        float* T = (l % 2 == 0) ? buf1 : buf2;   // GEMM output
        float* A = (l % 2 == 0) ? buf2 : buf1;   // aggregation accumulator
        gcn_gemm_wmma<<<gemm_blocks, 128, 0, stream>>>(H, Wm[l], T);
        self_init   <<<(NT + 255) / 256, b256, 0, stream>>>(T, dinv, A);
        edge_scatter<<<eblk, b256, 0, stream>>>(T, dinv, esrc, edst, A);
        if (l < 3) {
            bias_relu<<<(NT + 255) / 256, b256, 0, stream>>>(A, bm[l], T); // reuse T as H'
            H = T;
        } else {
            pool_add <<<(NT + 255) / 256, b256, 0, stream>>>(A, bm[l], batch, out);
        }
    }
}